// TorchKFBLock_22153441312983
// MI455X (gfx1250) — compile-verified
//
#include <hip/hip_runtime.h>

// ---------------------------------------------------------------------------
// 3-state Kalman filter scan, N = 1,048,576 steps, parallelized by chunking:
//   Phase 0: build R from guess/sign (exp10).
//   Phase 1: 4096 chunks x 256 steps, one chunk per lane. Covariance chain
//            re-seeded with a 64-step warm-up (Riccati map is contractive).
//            mCov stream is double-buffered through LDS with
//            GLOBAL_LOAD_ASYNC_TO_LDS_B128/B32 (ASYNCcnt / s_wait_asynccnt).
//            Each lane emits the chunk's composed affine map (A,b) for x.
//   Phase 2: hierarchical scan of the 4096 (A,b) summaries using
//            V_WMMA_F32_16X16X4_F32: one WMMA per composition,
//            D = A_k * [Acc|bcc] + [0|b_k]; composite kept in B-layout regs;
//            summary loads scalarized via readfirstlane (s_load), lane
//            placement via branch-free selects.
//   Phase 3: replay each chunk with its true start state (same async-LDS
//            staging), write out[N,3].
// All traffic (~80 MB) is L2-resident on MI455X (192 MB L2); streaming cost
// ~7 us at 23.3 TB/s, per-lane dependency chains are ~256+64 Riccati steps.
// ---------------------------------------------------------------------------

#define NSTEPS  1048576
#define CHUNK   256
#define NCHUNK  (NSTEPS / CHUNK)   // 4096
#define WARMUP  64
#define GROUP   128                // chunks per wave in the scan kernel
#define NWAVE   (NCHUNK / GROUP)   // 32 waves -> one 1024-thread workgroup

#define TSTEP   4                  // steps per async-staged tile
#define NTILE   (CHUNK / TSTEP)    // 64 tiles per chunk
#define ROWF    9                  // floats per mCov row
#define WPB     8                  // waves per 256-thread block

typedef __attribute__((ext_vector_type(2))) float v2f;
typedef __attribute__((ext_vector_type(8))) float v8f;

// ---------------- small dense helpers (row-major 3x3) ----------------------

__device__ __forceinline__ void inv3(const float* M, float* O) {
  float a = M[0], b = M[1], c = M[2];
  float d = M[3], e = M[4], f = M[5];
  float g = M[6], h = M[7], i = M[8];
  float A = e * i - f * h, B = c * h - b * i, C = b * f - c * e;
  float D = f * g - d * i, E = a * i - c * g, F = c * d - a * f;
  float G = d * h - e * g, H = b * g - a * h, I = a * e - b * d;
  float det = a * A + b * D + c * G;
  float r = 1.0f / det;
  O[0] = A * r; O[1] = B * r; O[2] = C * r;
  O[3] = D * r; O[4] = E * r; O[5] = F * r;
  O[6] = G * r; O[7] = H * r; O[8] = I * r;
}

__device__ __forceinline__ void mul33(const float* A, const float* B, float* C) {
#pragma unroll
  for (int r = 0; r < 3; ++r)
#pragma unroll
    for (int c = 0; c < 3; ++c)
      C[r * 3 + c] = A[r * 3 + 0] * B[0 * 3 + c]
                   + A[r * 3 + 1] * B[1 * 3 + c]
                   + A[r * 3 + 2] * B[2 * 3 + c];
}

__device__ __forceinline__ void mv3(const float* A, const float* x, float* y) {
#pragma unroll
  for (int r = 0; r < 3; ++r)
    y[r] = A[r * 3 + 0] * x[0] + A[r * 3 + 1] * x[1] + A[r * 3 + 2] * x[2];
}

__device__ __forceinline__ void cov_step(float* P, const float* R,
                                         const float* Mc, float* K) {
  float pr[9], S[9], Si[9], KP[9];
#pragma unroll
  for (int i = 0; i < 9; ++i) { pr[i] = P[i] + R[i]; S[i] = pr[i] + Mc[i]; }
  inv3(S, Si);
  mul33(pr, Si, K);
  mul33(K, pr, KP);
#pragma unroll
  for (int i = 0; i < 9; ++i) P[i] = pr[i] - KP[i];
}

__device__ __forceinline__ void load9(const float* p, float* d) {
#pragma unroll
  for (int i = 0; i < 9; ++i) d[i] = p[i];
}

// ---------------- async-to-LDS staging of the mCov stream -------------------
// Per-lane: TSTEP rows of 36 B, contiguous in both global memory and the
// lane's private LDS region, so one VADDR/VDST pair serves all 12 copies via
// immediate offsets (INST_OFFSET is added on both the LDS and global side).

__device__ __forceinline__ void stage_issue(unsigned ldsaddr,
                                            unsigned long long gaddr) {
#define CP128(OFF) asm volatile(                                             \
    "global_load_async_to_lds_b128 %0, %1, off offset:" OFF                  \
    :: "v"(ldsaddr), "v"(gaddr) : "memory")
#define CP32(OFF) asm volatile(                                              \
    "global_load_async_to_lds_b32 %0, %1, off offset:" OFF                   \
    :: "v"(ldsaddr), "v"(gaddr) : "memory")
  CP128("0");   CP128("16");  CP32("32");     // row 0
  CP128("36");  CP128("52");  CP32("68");     // row 1
  CP128("72");  CP128("88");  CP32("104");    // row 2
  CP128("108"); CP128("124"); CP32("140");    // row 3
#undef CP128
#undef CP32
}

__device__ __forceinline__ void wait_async_prev() {    // previous tile done
  asm volatile("s_wait_asynccnt 0xc" ::: "memory");    // 12 newest in flight
}
__device__ __forceinline__ void wait_async_all() {
  asm volatile("s_wait_asynccnt 0x0" ::: "memory");
}

// ---------------- Phase 0: build R -----------------------------------------

__global__ void kf_build_R(const float* __restrict__ guess,
                           const float* __restrict__ sgn,
                           float* __restrict__ Rout) {
  if (blockIdx.x == 0 && threadIdx.x == 0) {
    float d0 = exp10f(guess[0]), d1 = exp10f(guess[1]), d2 = exp10f(guess[2]);
    float o0 = sgn[0] * exp10f(guess[3]);
    float o1 = sgn[1] * exp10f(guess[4]);
    float o2 = sgn[2] * exp10f(guess[5]);
    Rout[0] = d0; Rout[1] = o0; Rout[2] = o1;
    Rout[3] = o0; Rout[4] = d1; Rout[5] = o2;
    Rout[6] = o1; Rout[7] = o2; Rout[8] = d2;
  }
}

// ---------------- Phase 1: per-chunk affine summaries (one chunk per lane) --

__global__ __launch_bounds__(256) void
kf_chunk_summary(const float* __restrict__ dt,
                 const float* __restrict__ prSig,
                 const float* __restrict__ mSig,
                 const float* __restrict__ mCov,
                 const float* __restrict__ R9,
                 float* __restrict__ chunkSum) {
  __shared__ float stage[2][WPB][32 * TSTEP * ROWF];
  const int lane = threadIdx.x & 31;
  const int wave = threadIdx.x >> 5;
  const int c = blockIdx.x * blockDim.x + threadIdx.x;   // grid == NCHUNK

  float R[9];
  load9(R9, R);

  const int kb = c * CHUNK;
  float P[9];
#pragma unroll
  for (int i = 0; i < 9; ++i) P[i] = 0.f;
  float K[9];

  // covariance warm-up (contractive Riccati map; exact for chunk 0)
  int kw = kb - WARMUP; if (kw < 1) kw = 1;
  for (int k = kw; k < kb; ++k) {
    __builtin_prefetch(mCov + (size_t)(k + 16) * 9, 0, 1);
    float Mc[9];
    load9(mCov + (size_t)k * 9, Mc);
    cov_step(P, R, Mc, K);
  }

  float* tile0 = &stage[0][wave][lane * TSTEP * ROWF];
  float* tile1 = &stage[1][wave][lane * TSTEP * ROWF];
  const unsigned l0 = (unsigned)(uintptr_t)tile0;
  const unsigned l1 = (unsigned)(uintptr_t)tile1;
  const float* gchunk = mCov + (size_t)kb * ROWF;

  float Acc[9] = {1.f, 0.f, 0.f, 0.f, 1.f, 0.f, 0.f, 0.f, 1.f};
  float bcc[3] = {0.f, 0.f, 0.f};

  stage_issue(l0, (unsigned long long)(uintptr_t)gchunk);
  for (int t = 0; t < NTILE; ++t) {
    const int buf = t & 1;
    if (t + 1 < NTILE) {
      stage_issue(buf ? l0 : l1,
                  (unsigned long long)(uintptr_t)(gchunk + (size_t)(t + 1) * TSTEP * ROWF));
      wait_async_prev();
    } else {
      wait_async_all();
    }
    const float* tp = buf ? tile1 : tile0;
#pragma unroll
    for (int j = 0; j < TSTEP; ++j) {
      const int k = kb + t * TSTEP + j;
      if (k >= 1) {                               // step 0 leaves state at 0
        float Mc[9];
        load9(tp + j * ROWF, Mc);
        cov_step(P, R, Mc, K);
        float Ak[9];
#pragma unroll
        for (int i = 0; i < 9; ++i) Ak[i] = -K[i];
        Ak[0] += 1.f; Ak[4] += 1.f; Ak[8] += 1.f;  // Ak = I - K
        float dtk = dt[k];
        float pv[3] = { prSig[(size_t)k * 3 + 0] * dtk,
                        prSig[(size_t)k * 3 + 1] * dtk,
                        prSig[(size_t)k * 3 + 2] * dtk };
        float ms[3] = { mSig[(size_t)k * 3 + 0],
                        mSig[(size_t)k * 3 + 1],
                        mSig[(size_t)k * 3 + 2] };
        float bk[3], u0[3], u1[3];
        mv3(Ak, pv, u0);
        mv3(K, ms, u1);
#pragma unroll
        for (int i = 0; i < 3; ++i) bk[i] = u0[i] + u1[i];
        float An[9];
        mul33(Ak, Acc, An);
#pragma unroll
        for (int i = 0; i < 9; ++i) Acc[i] = An[i];
        float bn[3];
        mv3(Ak, bcc, bn);
#pragma unroll
        for (int i = 0; i < 3; ++i) bcc[i] = bn[i] + bk[i];
      }
    }
  }

  float* s = chunkSum + (size_t)c * 12;
#pragma unroll
  for (int i = 0; i < 9; ++i) s[i] = Acc[i];
#pragma unroll
  for (int i = 0; i < 3; ++i) s[9 + i] = bcc[i];
}

// ---------------- Phase 2: WMMA-based hierarchical scan ---------------------
// Composite [Acc|bcc] held in V_WMMA B-operand layout (4x16 f32, 2 VGPRs):
// element (k,n) at vgpr = k&1, lane = n + 16*(k>>1) (K split across lane
// halves, mirroring the A layout). One composition per v_wmma_f32_16x16x4_f32.

__device__ __forceinline__ void wmma_compose(float& bs0, float& bs1, int lane,
                                             float A00, float A01, float A02,
                                             float A10, float A11, float A12,
                                             float A20, float A21, float A22,
                                             float b0, float b1, float b2) {
  const bool lo = lane < 16;
  const int m = lane & 15;
  // branch-free lane placement (v_cndmask chains, no exec manipulation)
  float axlo = (m == 0) ? A00 : (m == 1) ? A10 : (m == 2) ? A20 : 0.f; // K=0
  float axhi = (m == 0) ? A02 : (m == 1) ? A12 : (m == 2) ? A22 : 0.f; // K=2
  float aylo = (m == 0) ? A01 : (m == 1) ? A11 : (m == 2) ? A21 : 0.f; // K=1
  v2f a; a.x = lo ? axlo : axhi; a.y = lo ? aylo : 0.f;
  v2f b; b.x = bs0; b.y = bs1;
  const bool l3 = (lane == 3);
  v8f cc = {};
  cc[0] = l3 ? b0 : 0.f;                 // C(m,3): vgpr m, lane 3
  cc[1] = l3 ? b1 : 0.f;
  cc[2] = l3 ? b2 : 0.f;
  v8f d = __builtin_amdgcn_wmma_f32_16x16x4_f32(
      false, a, false, b, (short)0, cc, false, false);
  // D(r,n) at vgpr r, lane n.  Re-marshal rows 0..2 into B layout:
  float d2s = __shfl(d[2], m, 32);
  bs0 = lo ? d[0] : d2s;                 // k=0 rows / k=2 rows
  bs1 = lo ? d[1] : 0.f;                 // k=1 rows / k=3 stays zero
}

__global__ __launch_bounds__(1024) void
kf_scan(const float* __restrict__ chunkSum, float* __restrict__ chunkStart) {
  __shared__ float gsum[NWAVE * 12];
  __shared__ float gstart[NWAVE * 3];
  const int tid  = threadIdx.x;
  const int lane = tid & 31;
  const int wave = tid >> 5;

  // ---- phase A: each wave WMMA-composes its 128 chunk summaries ----
  // Base made wave-uniform so the 48 B summary rows come in as s_load_b128.
  const float4* sb = (const float4*)(chunkSum) +
                     (size_t)__builtin_amdgcn_readfirstlane(wave * GROUP) * 3;
  float bs0 = (lane == 0 || lane == 18) ? 1.f : 0.f;   // identity [I|0]
  float bs1 = (lane == 1) ? 1.f : 0.f;
  for (int i = 0; i < GROUP; ++i) {
    float4 q0 = sb[i * 3 + 0];   // A00 A01 A02 A10
    float4 q1 = sb[i * 3 + 1];   // A11 A12 A20 A21
    float4 q2 = sb[i * 3 + 2];   // A22 b0  b1  b2
    wmma_compose(bs0, bs1, lane, q0.x, q0.y, q0.z, q0.w,
                 q1.x, q1.y, q1.z, q1.w, q2.x, q2.y, q2.z, q2.w);
  }
  // extract composite: r0 = bs0 lanes0-3, r1 = bs1 lanes0-3, r2 = bs0 16-19
  float* g = gsum + wave * 12;
  if (lane < 3)                g[0 * 3 + lane]        = bs0;
  if (lane == 3)               g[9 + 0]               = bs0;
  if (lane < 3)                g[1 * 3 + lane]        = bs1;
  if (lane == 3)               g[9 + 1]               = bs1;
  if (lane >= 16 && lane < 19) g[2 * 3 + (lane - 16)] = bs0;
  if (lane == 19)              g[9 + 2]               = bs0;
  __syncthreads();

  // ---- phase B: wave 0 WMMA-scans the 32 group summaries ----
  if (wave == 0) {
    float t0 = (lane == 0 || lane == 18) ? 1.f : 0.f;
    float t1 = (lane == 1) ? 1.f : 0.f;
    for (int gi = 0; gi < NWAVE; ++gi) {
      float b0 = __shfl(t0, 3, 32);   // b of exclusive prefix = group start
      float b1 = __shfl(t1, 3, 32);
      float b2 = __shfl(t0, 19, 32);
      if (lane == 0) {
        gstart[gi * 3 + 0] = b0;
        gstart[gi * 3 + 1] = b1;
        gstart[gi * 3 + 2] = b2;
      }
      const float* s = gsum + gi * 12;
      wmma_compose(t0, t1, lane, s[0], s[1], s[2], s[3], s[4], s[5],
                   s[6], s[7], s[8], s[9], s[10], s[11]);
    }
  }
  __syncthreads();

  // ---- phase C: each wave propagates start states through its chunks ----
  float x[3] = { gstart[wave * 3 + 0], gstart[wave * 3 + 1], gstart[wave * 3 + 2] };
  const int cb = __builtin_amdgcn_readfirstlane(wave * GROUP);
  for (int i = 0; i < GROUP; ++i) {
    const int c = cb + i;
    if (lane == 0) {
      chunkStart[(size_t)c * 3 + 0] = x[0];
      chunkStart[(size_t)c * 3 + 1] = x[1];
      chunkStart[(size_t)c * 3 + 2] = x[2];
    }
    float4 q0 = sb[i * 3 + 0], q1 = sb[i * 3 + 1], q2 = sb[i * 3 + 2];
    float y0 = q0.x * x[0] + q0.y * x[1] + q0.z * x[2] + q2.y;
    float y1 = q0.w * x[0] + q1.x * x[1] + q1.y * x[2] + q2.z;
    float y2 = q1.z * x[0] + q1.w * x[1] + q2.x * x[2] + q2.w;
    x[0] = y0; x[1] = y1; x[2] = y2;
  }
}

// ---------------- Phase 3: replay with true start states --------------------

__global__ __launch_bounds__(256) void
kf_replay(const float* __restrict__ dt,
          const float* __restrict__ prSig,
          const float* __restrict__ mSig,
          const float* __restrict__ mCov,
          const float* __restrict__ R9,
          const float* __restrict__ chunkStart,
          float* __restrict__ out) {
  __shared__ float stage[2][WPB][32 * TSTEP * ROWF];
  const int lane = threadIdx.x & 31;
  const int wave = threadIdx.x >> 5;
  const int c = blockIdx.x * blockDim.x + threadIdx.x;   // grid == NCHUNK

  float R[9];
  load9(R9, R);

  const int kb = c * CHUNK;
  float P[9];
#pragma unroll
  for (int i = 0; i < 9; ++i) P[i] = 0.f;
  float K[9];

  int kw = kb - WARMUP; if (kw < 1) kw = 1;
  for (int k = kw; k < kb; ++k) {
    __builtin_prefetch(mCov + (size_t)(k + 16) * 9, 0, 1);
    float Mc[9];
    load9(mCov + (size_t)k * 9, Mc);
    cov_step(P, R, Mc, K);
  }

  float x[3] = { chunkStart[(size_t)c * 3 + 0],
                 chunkStart[(size_t)c * 3 + 1],
                 chunkStart[(size_t)c * 3 + 2] };
  if (kb == 0) { out[0] = 0.f; out[1] = 0.f; out[2] = 0.f; }  // state[0] = 0

  float* tile0 = &stage[0][wave][lane * TSTEP * ROWF];
  float* tile1 = &stage[1][wave][lane * TSTEP * ROWF];
  const unsigned l0 = (unsigned)(uintptr_t)tile0;
  const unsigned l1 = (unsigned)(uintptr_t)tile1;
  const float* gchunk = mCov + (size_t)kb * ROWF;

  stage_issue(l0, (unsigned long long)(uintptr_t)gchunk);
  for (int t = 0; t < NTILE; ++t) {
    const int buf = t & 1;
    if (t + 1 < NTILE) {
      stage_issue(buf ? l0 : l1,
                  (unsigned long long)(uintptr_t)(gchunk + (size_t)(t + 1) * TSTEP * ROWF));
      wait_async_prev();
    } else {
      wait_async_all();
    }
    const float* tp = buf ? tile1 : tile0;
#pragma unroll
    for (int j = 0; j < TSTEP; ++j) {
      const int k = kb + t * TSTEP + j;
      if (k >= 1) {
        float Mc[9];
        load9(tp + j * ROWF, Mc);
        cov_step(P, R, Mc, K);
        float dtk = dt[k];
        float px[3] = { x[0] + prSig[(size_t)k * 3 + 0] * dtk,
                        x[1] + prSig[(size_t)k * 3 + 1] * dtk,
                        x[2] + prSig[(size_t)k * 3 + 2] * dtk };
        float r0 = mSig[(size_t)k * 3 + 0] - px[0];
        float r1 = mSig[(size_t)k * 3 + 1] - px[1];
        float r2 = mSig[(size_t)k * 3 + 2] - px[2];
#pragma unroll
        for (int r = 0; r < 3; ++r)
          x[r] = px[r] + K[r * 3 + 0] * r0 + K[r * 3 + 1] * r1 + K[r * 3 + 2] * r2;
        out[(size_t)k * 3 + 0] = x[0];
        out[(size_t)k * 3 + 1] = x[1];
        out[(size_t)k * 3 + 2] = x[2];
      }
    }
  }
}

// ---------------------------------------------------------------------------

extern "C" void kernel_launch(void* const* d_in, const int* in_sizes, int n_in,
                              void* d_out, int out_size, void* d_ws, size_t ws_size,
                              hipStream_t stream) {
  (void)in_sizes; (void)n_in; (void)out_size; (void)ws_size;
  const float* guess = (const float*)d_in[0];
  const float* sgn   = (const float*)d_in[1];
  const float* dt    = (const float*)d_in[2];
  const float* prSig = (const float*)d_in[3];
  const float* mSig  = (const float*)d_in[4];
  const float* mCov  = (const float*)d_in[5];
  float* out = (float*)d_out;

  float* ws         = (float*)d_ws;
  float* R          = ws;                         // 9 (padded to 16)
  float* chunkSum   = ws + 16;                    // 12 * NCHUNK floats
  float* chunkStart = ws + 16 + 12 * NCHUNK;      // 3 * NCHUNK floats

  kf_build_R<<<1, 32, 0, stream>>>(guess, sgn, R);
  kf_chunk_summary<<<NCHUNK / 256, 256, 0, stream>>>(dt, prSig, mSig, mCov, R, chunkSum);
  kf_scan<<<1, 1024, 0, stream>>>(chunkSum, chunkStart);
  kf_replay<<<NCHUNK / 256, 256, 0, stream>>>(dt, prSig, mSig, mCov, R, chunkStart, out);
}